// HashEmbedder_90220083020383
// MI455X (gfx1250) — compile-verified
//
#include <hip/hip_runtime.h>
#include <stdint.h>

// ---------------------------------------------------------------------------
// HashEmbedder (instant-NGP 2D hash grid), MI455X-tuned.
// Memory-bound: ~200MB HBM traffic -> ~9us floor @ 23.3 TB/s.
// Strategy: 8 threads/point (2 levels each), fully coalesced b128 NT stores,
// 8 outstanding b64 gathers per thread (tables resident in 192MB L2).
// ---------------------------------------------------------------------------

#define LOG2_T      19
#define TABLE_SIZE  (1u << LOG2_T)          // 524288 entries
#define HASH_MASK   (TABLE_SIZE - 1u)
#define PRIME1      2654435761u
#define N_LEVELS    16

// native vector type acceptable to __builtin_nontemporal_store
typedef float v4f __attribute__((ext_vector_type(4)));

// res[l] = floor(16 * (512/16)^(l/15)) = floor(16 * 2^(l/3))
__device__ __constant__ float c_res[N_LEVELS] = {
    16.f, 20.f, 25.f, 32.f, 40.f, 50.f, 64.f, 80.f,
    101.f, 128.f, 161.f, 203.f, 256.f, 322.f, 406.f, 512.f
};

__global__ __launch_bounds__(256) void HashEmbedder_90220083020383_kernel(
    const float* __restrict__ x,          // [n,2]
    const float* __restrict__ emb,        // [16, 2^19, 2]
    float* __restrict__ out,              // [n,32]
    int n)
{
    const uint32_t g     = blockIdx.x * 256u + threadIdx.x;
    const uint32_t point = g >> 3;        // 8 threads per point
    const uint32_t sub   = g & 7u;        // which level-pair (2 levels each)
    if (point >= (uint32_t)n) return;

    // 8 lanes hit the same 8B -> single L0 line, broadcast.
    const float2 p = *(const float2*)(x + 2u * (size_t)point);

    float acc[4];

#pragma unroll
    for (int k = 0; k < 2; ++k) {
        const int   level = (int)(sub << 1) + k;
        const float res   = c_res[level];
        const int   resi  = (int)res;

        const float px = p.x * res;
        const float py = p.y * res;
        const float fx = floorf(px);
        const float fy = floorf(py);

        int x0 = (int)fx;
        int y0 = (int)fy;
        // reference clips box to [0, res] (inclusive upper)
        int x1 = min(x0 + 1, resi);
        int y1 = min(y0 + 1, resi);
        x0 = min(max(x0, 0), resi);
        y0 = min(max(y0, 0), resi);

        const float wx = px - fx;
        const float wy = py - fy;

        const uint32_t hy0 = (uint32_t)y0 * PRIME1;
        const uint32_t hy1 = (uint32_t)y1 * PRIME1;
        const uint32_t h00 = ((uint32_t)x0 ^ hy0) & HASH_MASK;
        const uint32_t h10 = ((uint32_t)x1 ^ hy0) & HASH_MASK;
        const uint32_t h01 = ((uint32_t)x0 ^ hy1) & HASH_MASK;
        const uint32_t h11 = ((uint32_t)x1 ^ hy1) & HASH_MASK;

        const float* tbl = emb + (size_t)level * (TABLE_SIZE * 2u);

        // four b64 gathers; L2-resident after first touch (64MB tables, 192MB L2)
        const float2 f00 = *(const float2*)(tbl + 2u * h00);
        const float2 f10 = *(const float2*)(tbl + 2u * h10);
        const float2 f01 = *(const float2*)(tbl + 2u * h01);
        const float2 f11 = *(const float2*)(tbl + 2u * h11);

        const float w00 = (1.0f - wx) * (1.0f - wy);
        const float w10 = wx * (1.0f - wy);
        const float w01 = (1.0f - wx) * wy;
        const float w11 = wx * wy;

        acc[2 * k + 0] = f00.x * w00 + f10.x * w10 + f01.x * w01 + f11.x * w11;
        acc[2 * k + 1] = f00.y * w00 + f10.y * w10 + f01.y * w01 + f11.y * w11;
    }

    // out[point, sub*4 .. sub*4+3]  ==  out + g*4: consecutive lanes write
    // consecutive float4 -> wave32 stores 512B contiguous per instruction.
    // Non-temporal: keep the 128MB output stream from evicting tables in L2.
    v4f r;
    r.x = acc[0]; r.y = acc[1]; r.z = acc[2]; r.w = acc[3];
    __builtin_nontemporal_store(r, (v4f*)(out + (size_t)g * 4u));
}

extern "C" void kernel_launch(void* const* d_in, const int* in_sizes, int n_in,
                              void* d_out, int out_size, void* d_ws, size_t ws_size,
                              hipStream_t stream) {
    (void)n_in; (void)out_size; (void)d_ws; (void)ws_size;
    const float* x   = (const float*)d_in[0];   // [n,2] fp32
    const float* emb = (const float*)d_in[1];   // [16, 2^19, 2] fp32
    float* out       = (float*)d_out;           // [n,32] fp32

    const int n = in_sizes[0] / 2;
    const long long total = 8LL * n;            // 8 threads per point
    const int block = 256;
    const int grid  = (int)((total + block - 1) / block);

    HashEmbedder_90220083020383_kernel<<<grid, block, 0, stream>>>(x, emb, out, n);
}